// PredictsEnsemble_83665962926803
// MI455X (gfx1250) — compile-verified
//
#include <hip/hip_runtime.h>
#include <cstdint>

// Problem constants
#define B_TOT 131072
#define M_N   2
#define C_N   219
#define BT    16      // b-rows per block
#define TPB   256     // 8 waves (wave32)

typedef unsigned int uint4v __attribute__((ext_vector_type(4)));
typedef int          int8v  __attribute__((ext_vector_type(8)));
typedef int          int4v  __attribute__((ext_vector_type(4)));

// Build a D# (groups 0/1, groups 2/3 zero => 2D max) and issue TENSOR_LOAD_TO_LDS.
// Field packing per CDNA5 ISA (verified against emitted SGPR values).
__device__ __forceinline__ void tdm_load_2d(uint32_t lds_byte_addr,
                                            uint64_t gaddr,
                                            uint32_t tensor_d0, uint32_t tensor_d1,
                                            uint32_t tile_d0,   uint32_t tile_d1,
                                            uint64_t stride0 /* dim0 stride, elems */)
{
    uint4v g0;
    g0.x = 1u;                                   // count=1, is_restore=0, gather=0
    g0.y = lds_byte_addr;                        // [63:32]  lds_addr (bytes)
    g0.z = (uint32_t)(gaddr & 0xFFFFFFFFu);      // [95:64]  global_addr lo
    g0.w = (uint32_t)((gaddr >> 32) & 0x1FFFFFFu)// [120:96] global_addr hi
         | (2u << 30);                           // [127:126] type = 2 ("image")

    int8v g1;
    g1[0] = (int)(2u << 16);                                     // data_size=4B; wg_mask=0
    g1[1] = (int)((tensor_d0 & 0xFFFFu) << 16);                  // atomic_bar_addr=0 | td0[15:0]
    g1[2] = (int)(((tensor_d0 >> 16) & 0xFFFFu) |
                  ((tensor_d1 & 0xFFFFu) << 16));                // td0[31:16] | td1[15:0]
    g1[3] = (int)(((tensor_d1 >> 16) & 0xFFFFu) |
                  ((tile_d0 & 0xFFFFu) << 16));                  // td1[31:16] | tile0
    g1[4] = (int)(tile_d1 & 0xFFFFu);                            // tile1 | tile2=0
    g1[5] = (int)(uint32_t)(stride0 & 0xFFFFFFFFu);              // stride0[31:0]
    g1[6] = (int)(uint32_t)((stride0 >> 32) & 0xFFFFu);          // stride0[47:32] | stride1[15:0]=0
    g1[7] = 0;                                                   // stride1[47:16]=0

    int4v z4 = {0, 0, 0, 0};
    int8v z8 = {0, 0, 0, 0, 0, 0, 0, 0};
    __builtin_amdgcn_tensor_load_to_lds(g0, g1, z4, z4, z8, 0);
}

__global__ __launch_bounds__(TPB)
void PredictsEnsemble_kernel(const float* __restrict__ pred,   // flat [B*M*C]
                             const float* __restrict__ lin_w,  // [C,2]
                             const float* __restrict__ lin_b,  // [C]
                             const float* __restrict__ conv_w, // [2]
                             float* __restrict__ out)          // [B,C]
{
    // Term-1 view tile: row c holds flat[c*2B + 2*b0 + j], j=0..2*BT-1  (128 B rows, 1 MB stride)
    __shared__ float lds_t1[C_N * 2 * BT];          // 28032 B
    // Term-2 view tile: contiguous chunk flat[b0*438 .. +BT*438)
    __shared__ float lds_t2[BT * M_N * C_N];        // 28032 B
    __shared__ float lds_w0[C_N], lds_w1[C_N], lds_bb[C_N];

    const uint32_t tid = threadIdx.x;
    const uint32_t b0  = blockIdx.x * BT;

    if ((tid >> 5) == 0) {   // one wave issues the DMAs (branch confirmed: s_cbranch_execz)
        // 1D bulk copy: BT*438 contiguous floats of the [B,M,C] view
        tdm_load_2d((uint32_t)(uintptr_t)&lds_t2[0],
                    (uint64_t)(uintptr_t)pred + (uint64_t)b0 * (M_N * C_N) * 4ull,
                    0x7FFFFFFFu, 1u,
                    (uint32_t)(BT * M_N * C_N), 0u,
                    (uint64_t)(BT * M_N * C_N));
        // 2D strided gather: C_N rows x (2*BT) floats, row stride = 2*B elements
        tdm_load_2d((uint32_t)(uintptr_t)&lds_t1[0],
                    (uint64_t)(uintptr_t)pred + (uint64_t)b0 * 2ull * 4ull,
                    (uint32_t)(2 * B_TOT), (uint32_t)C_N,
                    (uint32_t)(2 * BT),    (uint32_t)C_N,
                    (uint64_t)(2 * B_TOT));
    }

    // Per-class weights/bias -> LDS (tiny, cached loads)
    if (tid < C_N) {
        lds_bb[tid] = lin_b[tid];
        lds_w0[tid] = lin_w[2 * tid + 0];
        lds_w1[tid] = lin_w[2 * tid + 1];
    }
    const float cw0 = conv_w[0];
    const float cw1 = conv_w[1];

    __builtin_amdgcn_s_wait_tensorcnt(0);   // DMA done (no-op for waves that issued nothing)
    __syncthreads();

    // Each block writes one contiguous span out[b0*219 .. +BT*219): fully coalesced.
    // NT stores: output is write-once / never re-read; keep L2 capacity for the
    // twice-read 229.6 MB input instead.
    float* __restrict__ outrow = out + (size_t)b0 * C_N;
    #pragma unroll 2
    for (uint32_t idx = tid; idx < BT * C_N; idx += TPB) {
        const uint32_t brel = idx / C_N;            // const-div -> mul/shift
        const uint32_t c    = idx - brel * C_N;

        const float a0 = lds_t1[c * (2 * BT) + 2 * brel + 0];
        const float a1 = lds_t1[c * (2 * BT) + 2 * brel + 1];
        float v1 = fmaf(a0, lds_w0[c], fmaf(a1, lds_w1[c], lds_bb[c]));
        v1 = fmaxf(v1, 0.0f);                       // relu

        const float p0 = lds_t2[brel * (M_N * C_N) + c];
        const float p1 = lds_t2[brel * (M_N * C_N) + C_N + c];
        const float v2 = fmaf(p0, cw0, p1 * cw1);

        __builtin_nontemporal_store(v1 + v2, &outrow[idx]);
    }
}

extern "C" void kernel_launch(void* const* d_in, const int* in_sizes, int n_in,
                              void* d_out, int out_size, void* d_ws, size_t ws_size,
                              hipStream_t stream) {
    (void)in_sizes; (void)n_in; (void)out_size; (void)d_ws; (void)ws_size;
    const float* pred   = (const float*)d_in[0];
    const float* lin_w  = (const float*)d_in[1];
    const float* lin_b  = (const float*)d_in[2];
    const float* conv_w = (const float*)d_in[3];
    float* out = (float*)d_out;

    dim3 grid(B_TOT / BT);   // 8192 blocks
    PredictsEnsemble_kernel<<<grid, TPB, 0, stream>>>(pred, lin_w, lin_b, conv_w, out);
}